// RotmatQCQPSolver_54949811585160
// MI455X (gfx1250) — compile-verified
//
#include <hip/hip_runtime.h>

typedef float v2f __attribute__((ext_vector_type(2)));
typedef float v4f __attribute__((ext_vector_type(4)));
typedef float v8f __attribute__((ext_vector_type(8)));

#if defined(__has_builtin)
#if __has_builtin(__builtin_amdgcn_wmma_f32_16x16x4_f32)
#define USE_WMMA_F32 1
#endif
#endif

namespace {

constexpr int NX      = 10;           // state dim (x = [vec(R); 1])
constexpr int NC      = 22;           // constraint count
constexpr int GSTRIDE = 12;           // padded row stride of G^T (10 + 2 zero pad)
constexpr int GT_ROWS = 32;           // rows 0..21 G, 22 = Ax, 23..31 zero pad
constexpr int L_OFF   = GT_ROWS * GSTRIDE;            // 384: lower-tri gram
constexpr int R_OFF   = L_OFF + (NC * (NC + 1)) / 2;  // 637: rhs (G^T Ax)
constexpr int DUMP    = R_OFF + NC;                   // 659: dump dword
constexpr int SLOT    = 676;          // dwords/item: 16B aligned, stride 36 mod 64
constexpr int NITERS  = 15;
constexpr int NSWEEP  = 8;            // cyclic Jacobi sweeps

constexpr int LT(int i, int j) { return i * (i + 1) / 2 + j; }

// ---- constraint matrices, mirrors build_constraints() exactly ----
struct ETab { float v[NC][NX][NX]; };

constexpr ETab build_E() {
  ETab t{};
  int k = 0;
  for (int a = 0; a < 3; ++a)
    for (int b = a; b < 3; ++b) {
      for (int i = 0; i < 3; ++i) {
        t.v[k][3 * a + i][3 * b + i] += 0.5f;
        t.v[k][3 * b + i][3 * a + i] += 0.5f;
      }
      ++k;
    }
  t.v[k][9][9] = 1.0f;
  ++k;
  for (int i = 0; i < 3; ++i)
    for (int j = i; j < 3; ++j) {
      for (int a = 0; a < 3; ++a) {
        t.v[k][3 * a + i][3 * a + j] += 0.5f;
        t.v[k][3 * a + j][3 * a + i] += 0.5f;
      }
      ++k;
    }
  const int abc[3][3] = {{0, 1, 2}, {1, 2, 0}, {2, 0, 1}};
  for (int q = 0; q < 3; ++q) {
    const int a = abc[q][0], b = abc[q][1], cc = abc[q][2];
    for (int i = 0; i < 3; ++i) {
      const int j = (i + 1) % 3, l = (i + 2) % 3;
      t.v[k][3 * a + j][3 * b + l] += 0.5f;
      t.v[k][3 * b + l][3 * a + j] += 0.5f;
      t.v[k][3 * a + l][3 * b + j] -= 0.5f;
      t.v[k][3 * b + j][3 * a + l] -= 0.5f;
      t.v[k][3 * cc + i][9] -= 0.5f;
      t.v[k][9][3 * cc + i] -= 0.5f;
      ++k;
    }
  }
  return t;
}

constexpr ETab E_TAB = build_E();

} // namespace

// One item per thread; one wave (32 items) per workgroup so the wave can
// cooperatively run the WMMA gram stage over its lanes' items via LDS.
__global__ __launch_bounds__(32) void rotmat_qcqp_kernel(
    const float* __restrict__ A_vec, float* __restrict__ out, int nitems) {
  extern __shared__ float lds[];
  const int lane = threadIdx.x;                // 0..31, wave32
  int item = blockIdx.x * 32 + lane;
  if (item >= nitems) item = nitems - 1;       // clamp: keeps EXEC all-ones
  float* slot = lds + lane * SLOT;             // this lane's item slot

  // ---- load A (55 upper-tri) -> dense symmetric 10x10 in registers ----
  float A[NX][NX];
  {
    const float* av = A_vec + (size_t)item * 55;
    int idx = 0;
#pragma unroll
    for (int i = 0; i < NX; ++i)
#pragma unroll
      for (int j = i; j < NX; ++j) {
        const float v = av[idx++];
        A[i][j] = v;
        A[j][i] = v;
      }
  }

  float nu[NC];
#pragma unroll
  for (int k = 0; k < NC; ++k) nu[k] = 0.0f;

  float x[NX];

  for (int it = 0; it <= NITERS; ++it) {
    // ---- M = A + sum_k nu_k * E_k  (E_k constexpr sparse; zeros fold) ----
    float M[NX][NX];
#pragma unroll
    for (int m = 0; m < NX; ++m)
#pragma unroll
      for (int n = 0; n < NX; ++n) {
        float s = A[m][n];
#pragma unroll
        for (int k = 0; k < NC; ++k) {
          const float e = E_TAB.v[k][m][n];
          if (e != 0.0f) s = fmaf(nu[k], e, s);
        }
        M[m][n] = s;
      }

    // ---- cyclic Jacobi eigensolve (branch-free rotations), accumulate V ----
    float V[NX][NX];
#pragma unroll
    for (int i = 0; i < NX; ++i)
#pragma unroll
      for (int j = 0; j < NX; ++j) V[i][j] = (i == j) ? 1.0f : 0.0f;

    for (int sweep = 0; sweep < NSWEEP; ++sweep) {
#pragma unroll
      for (int p = 0; p < NX - 1; ++p)
#pragma unroll
        for (int q = p + 1; q < NX; ++q) {
          const float apq = M[p][q];
          const bool rot = fabsf(apq) > 1e-20f;
          const float den = rot ? (2.0f * apq) : 1.0f;   // safe denominator
          const float theta = (M[q][q] - M[p][p]) / den;
          float tt = 1.0f / (fabsf(theta) + sqrtf(fmaf(theta, theta, 1.0f)));
          tt = (theta < 0.0f) ? -tt : tt;
          tt = rot ? tt : 0.0f;                          // identity if tiny
          const float c = 1.0f / sqrtf(fmaf(tt, tt, 1.0f));
          const float s = tt * c;
#pragma unroll
          for (int r = 0; r < NX; ++r) {   // M <- M * J
            const float mrp = M[r][p], mrq = M[r][q];
            M[r][p] = fmaf(c, mrp, -s * mrq);
            M[r][q] = fmaf(s, mrp, c * mrq);
          }
#pragma unroll
          for (int r = 0; r < NX; ++r) {   // M <- J^T * M
            const float mpr = M[p][r], mqr = M[q][r];
            M[p][r] = fmaf(c, mpr, -s * mqr);
            M[q][r] = fmaf(s, mpr, c * mqr);
          }
#pragma unroll
          for (int r = 0; r < NX; ++r) {   // V <- V * J
            const float vrp = V[r][p], vrq = V[r][q];
            V[r][p] = fmaf(c, vrp, -s * vrq);
            V[r][q] = fmaf(s, vrp, c * vrq);
          }
        }
    }

    // ---- min eigenvector, normalize to x[9] = 1 ----
    float bestv = M[0][0];
    int bi = 0;
#pragma unroll
    for (int d = 1; d < NX; ++d)
      if (M[d][d] < bestv) { bestv = M[d][d]; bi = d; }
#pragma unroll
    for (int r = 0; r < NX; ++r) {
      float xv = V[r][0];
#pragma unroll
      for (int d = 1; d < NX; ++d) xv = (bi == d) ? V[r][d] : xv;
      x[r] = xv;
    }
    {
      const float h = x[9];
      const float denom = (fabsf(h) < 1e-6f) ? ((h >= 0.0f) ? 1e-6f : -1e-6f) : h;
      const float inv = 1.0f / denom;
#pragma unroll
      for (int r = 0; r < NX; ++r) x[r] *= inv;
    }

    if (it == NITERS) break;

    // ---- Ax ----
    float Ax[NX];
#pragma unroll
    for (int m = 0; m < NX; ++m) {
      float s = 0.0f;
#pragma unroll
      for (int n = 0; n < NX; ++n) s = fmaf(A[m][n], x[n], s);
      Ax[m] = s;
    }

    // ---- G^T (zero padded) into this lane's LDS slot: row i holds (E_i x),
    //      row 22 holds Ax, rows 23..31 and cols 10..11 are zeros ----
#pragma unroll
    for (int i = 0; i < NC; ++i) {
      float g[NX];
#pragma unroll
      for (int m = 0; m < NX; ++m) {
        float s = 0.0f;
#pragma unroll
        for (int n = 0; n < NX; ++n) {
          const float e = E_TAB.v[i][m][n];
          if (e != 0.0f) s = fmaf(e, x[n], s);
        }
        g[m] = s;
      }
      v4f* dst = (v4f*)(slot + i * GSTRIDE);
      v4f w0 = {g[0], g[1], g[2], g[3]};
      v4f w1 = {g[4], g[5], g[6], g[7]};
      v4f w2 = {g[8], g[9], 0.0f, 0.0f};
      dst[0] = w0; dst[1] = w1; dst[2] = w2;
    }
    {
      v4f* dst = (v4f*)(slot + NC * GSTRIDE);     // row 22 = Ax
      v4f a0 = {Ax[0], Ax[1], Ax[2], Ax[3]};
      v4f a1 = {Ax[4], Ax[5], Ax[6], Ax[7]};
      v4f a2 = {Ax[8], Ax[9], 0.0f, 0.0f};
      dst[0] = a0; dst[1] = a1; dst[2] = a2;
      const v4f z = {0.0f, 0.0f, 0.0f, 0.0f};
#pragma unroll
      for (int i = NC + 1; i < GT_ROWS; ++i) {    // rows 23..31 = 0
        v4f* zd = (v4f*)(slot + i * GSTRIDE);
        zd[0] = z; zd[1] = z; zd[2] = z;
      }
    }
    __syncthreads();

    // ---- Gram + rhs via V_WMMA_F32_16X16X4_F32, one item at a time:
    //      D(32x32, 2x2 tiles) = Gext^T (23x10) * Gext (10x23); all operand
    //      loads are unconditional b64 (zero padding makes OOB lanes read 0),
    //      scatter uses select-computed addresses (invalid -> dump dword) ----
    const int half = lane >> 4;           // 0|1
    const int lp = lane & 15;
    const int kofs = 2 * half;            // K offset inside a K=4 chunk
    for (int t = 0; t < 32; ++t) {
      float* gs = lds + t * SLOT;
#pragma unroll
      for (int ti = 0; ti < 2; ++ti) {
#pragma unroll
        for (int tj = 0; tj < 2; ++tj) {
#if defined(USE_WMMA_F32)
          v8f acc = {0.0f, 0.0f, 0.0f, 0.0f, 0.0f, 0.0f, 0.0f, 0.0f};
#pragma unroll
          for (int cc = 0; cc < 3; ++cc) {
            const int k0 = cc * 4 + kofs;
            // A 16x4: lane -> row, VGPR0/1 -> K{k0,k0+1}; B 4x16: lane -> col
            const v2f a = *(const v2f*)(gs + (ti * 16 + lp) * GSTRIDE + k0);
            const v2f b = *(const v2f*)(gs + (tj * 16 + lp) * GSTRIDE + k0);
            acc = __builtin_amdgcn_wmma_f32_16x16x4_f32(
                false, a, false, b, (short)0, acc, false, false);
          }
#pragma unroll
          for (int vv = 0; vv < 8; ++vv) {
            const int row = ti * 16 + vv + 8 * half;   // C/D layout
            const int col = tj * 16 + lp;
            const bool inL = (row < NC) && (col <= row);
            const bool inR = (row < NC) && (col == NC);
            const int aL = L_OFF + (row * (row + 1)) / 2 + col;
            const int addr = inL ? aL : (inR ? (R_OFF + row) : DUMP);
            gs[addr] = acc[vv];
          }
#else
          // exact VALU fallback (zero-padded rows make OOB products zero)
#pragma unroll
          for (int vv = 0; vv < 8; ++vv) {
            const int row = ti * 16 + vv + 8 * half;
            const int col = tj * 16 + lp;
            float val = 0.0f;
#pragma unroll
            for (int m = 0; m < NX; ++m)
              val = fmaf(gs[row * GSTRIDE + m], gs[col * GSTRIDE + m], val);
            const bool inL = (row < NC) && (col <= row);
            const bool inR = (row < NC) && (col == NC);
            const int aL = L_OFF + (row * (row + 1)) / 2 + col;
            const int addr = inL ? aL : (inR ? (R_OFF + row) : DUMP);
            gs[addr] = val;
          }
#endif
        }
      }
    }
    __syncthreads();

    // ---- per-lane Cholesky solve of (gram + 1e-6 I) nu = -G^T Ax ----
    float L[(NC * (NC + 1)) / 2];
#pragma unroll
    for (int i = 0; i < NC; ++i)
#pragma unroll
      for (int j = 0; j <= i; ++j) L[LT(i, j)] = slot[L_OFF + LT(i, j)];
#pragma unroll
    for (int i = 0; i < NC; ++i) L[LT(i, i)] += 1e-6f;
    float rhs[NC];
#pragma unroll
    for (int i = 0; i < NC; ++i) rhs[i] = -slot[R_OFF + i];

    float dinv[NC];
#pragma unroll
    for (int j = 0; j < NC; ++j) {
      float s = L[LT(j, j)];
#pragma unroll
      for (int t2 = 0; t2 < j; ++t2) s = fmaf(-L[LT(j, t2)], L[LT(j, t2)], s);
      const float d = sqrtf(s);
      L[LT(j, j)] = d;
      const float di = 1.0f / d;
      dinv[j] = di;
#pragma unroll
      for (int i = j + 1; i < NC; ++i) {
        float s2 = L[LT(i, j)];
#pragma unroll
        for (int t2 = 0; t2 < j; ++t2) s2 = fmaf(-L[LT(i, t2)], L[LT(j, t2)], s2);
        L[LT(i, j)] = s2 * di;
      }
    }
    float y[NC];
#pragma unroll
    for (int i = 0; i < NC; ++i) {
      float s = rhs[i];
#pragma unroll
      for (int j = 0; j < i; ++j) s = fmaf(-L[LT(i, j)], y[j], s);
      y[i] = s * dinv[i];
    }
#pragma unroll
    for (int i = NC - 1; i >= 0; --i) {
      float s = y[i];
#pragma unroll
      for (int j = i + 1; j < NC; ++j) s = fmaf(-L[LT(j, i)], nu[j], s);
      nu[i] = s * dinv[i];
    }
  }

  // ---- final x ----
#pragma unroll
  for (int r = 0; r < NX; ++r) out[(size_t)item * NX + r] = x[r];
}

extern "C" void kernel_launch(void* const* d_in, const int* in_sizes, int n_in,
                              void* d_out, int out_size, void* d_ws, size_t ws_size,
                              hipStream_t stream) {
  (void)n_in; (void)out_size; (void)d_ws; (void)ws_size;
  const float* A_vec = (const float*)d_in[0];
  float* out = (float*)d_out;
  const int nitems = in_sizes[0] / 55;             // 32768
  const int blocks = (nitems + 31) / 32;           // 1 wave (32 items) per block
  const size_t shmem = 32u * SLOT * sizeof(float); // 86.5 KB of 320 KB WGP LDS
  rotmat_qcqp_kernel<<<blocks, 32, shmem, stream>>>(A_vec, out, nitems);
}